// EdgeAttrHeteroConv_38250978738461
// MI455X (gfx1250) — compile-verified
//
#include <hip/hip_runtime.h>
#include <math.h>

// MI455X / gfx1250, wave32. fp32 WMMA (V_WMMA_F32_16X16X4_F32) for all GEMMs.

typedef __attribute__((ext_vector_type(2))) float v2f;
typedef __attribute__((ext_vector_type(8))) float v8f;

#define CCH 256   // channels

// ---------------------------------------------------------------------------
// GEMM: D[M,256] = act(A[M,256] @ B[256,256] + bias[256])
// act: 0 = identity, 1 = sigmoid
// Block = 128 threads (4 waves). B column panel [256x64] staged in LDS once,
// row-pair interleaved + XOR-32 bank swizzle, so every B fragment is one
// 8B-aligned ds_load_b64 that feeds v_wmma directly (no VGPR repacking).
// Each wave computes a 32x64 tile: 8 accumulators, 8 WMMAs per k-step of 4.
// A rows are CLAMPED (not predicated) so the inner loop is branch-free;
// out-of-range rows compute garbage that is never stored.
// ---------------------------------------------------------------------------
__global__ void __launch_bounds__(128)
wmma_gemm_bias(const float* __restrict__ A, const float* __restrict__ B,
               const float* __restrict__ bias, float* __restrict__ D,
               int M, int act)
{
    __shared__ float Bs[128 * 128];         // 64 KB: [row-pair][interleaved col]
    const int lane = threadIdx.x & 31;
    const int wave = threadIdx.x >> 5;
    const int n0 = blockIdx.y * 64;

    // cooperative stage: logical (r, c) -> Bs[(r>>1)*128 + ((2c) ^ ((rp&1)<<5))]
    for (int idx = threadIdx.x; idx < 128 * 64; idx += 128) {
        const int rp = idx >> 6, c = idx & 63;
        v2f v;
        v.x = B[(size_t)(2 * rp)     * CCH + n0 + c];
        v.y = B[(size_t)(2 * rp + 1) * CCH + n0 + c];
        const int pc = (2 * c) ^ ((rp & 1) << 5);
        *(v2f*)(&Bs[rp * 128 + pc]) = v;    // ds_store_b64
    }
    __syncthreads();

    const int m0 = (blockIdx.x * 4 + wave) * 32;
    if (m0 >= M) return;                    // wave-uniform, no barriers after

    const int mr = lane & 15;               // A row / B,C col within 16-tile
    const int kb = (lane >> 4) << 1;        // K sub-offset: 0 / 2
    const int r0 = min(m0 + mr, M - 1);     // clamped rows -> branch-free loop
    const int r1 = min(m0 + 16 + mr, M - 1);
    const float* A0 = A + (size_t)r0 * CCH;
    const float* A1 = A + (size_t)r1 * CCH;

    v8f acc[8] = {};                        // [mtile(2)][ntile(4)]
    for (int k = 0; k < CCH; k += 4) {
        __builtin_prefetch(A0 + k + 64, 0, 3);          // near-scope prefetch
        const v2f a0 = *(const v2f*)(A0 + k + kb);      // global_load_b64
        const v2f a1 = *(const v2f*)(A1 + k + kb);
        const int rp = (k + kb) >> 1;                   // row-pair index
        const float* bp = &Bs[rp * 128];
        const int sw = (rp & 1) << 5;                   // bank swizzle
#pragma unroll
        for (int t = 0; t < 4; ++t) {
            const v2f b = *(const v2f*)(bp + (((mr + 16 * t) * 2) ^ sw)); // ds_load_b64
            acc[t] = __builtin_amdgcn_wmma_f32_16x16x4_f32(
                false, a0, false, b, (short)0, acc[t], false, false);
            acc[4 + t] = __builtin_amdgcn_wmma_f32_16x16x4_f32(
                false, a1, false, b, (short)0, acc[4 + t], false, false);
        }
    }

    const int rbase = (lane >> 4) * 8;      // C/D: lanes16-31 hold M=r+8
#pragma unroll
    for (int mt = 0; mt < 2; ++mt) {
#pragma unroll
        for (int r = 0; r < 8; ++r) {
            const int row = m0 + mt * 16 + rbase + r;
            if (row < M) {
#pragma unroll
                for (int t = 0; t < 4; ++t) {
                    const int col = n0 + t * 16 + mr;
                    float v = acc[mt * 4 + t][r] + bias[col];
                    if (act == 1) v = 1.0f / (1.0f + expf(-v));
                    D[(size_t)row * CCH + col] = v;
                }
            }
        }
    }
}

// ---------------------------------------------------------------------------
// Tiny gate tables:
//   at_part[10,256] = at @ Wg_aff[0:32,:]
//   as_part[5,256]  = as @ Wg_aff[32:64,:]
//   ph_gate[3,256]  = sigmoid(ph @ Wg_ph + bg_ph)
// ---------------------------------------------------------------------------
__global__ void gate_tables(const float* __restrict__ at, const float* __restrict__ as_,
                            const float* __restrict__ ph,
                            const float* __restrict__ WgA, const float* __restrict__ WgP,
                            const float* __restrict__ bgP,
                            float* __restrict__ atp, float* __restrict__ asp,
                            float* __restrict__ phg)
{
    const int idx = blockIdx.x * blockDim.x + threadIdx.x;
    if (idx < 10 * 256) {
        const int r = idx >> 8, c = idx & 255;
        float s = 0.0f;
        for (int k = 0; k < 32; ++k) s += at[r * 32 + k] * WgA[k * 256 + c];
        atp[idx] = s;
    } else if (idx < 15 * 256) {
        const int j = idx - 10 * 256, r = j >> 8, c = j & 255;
        float s = 0.0f;
        for (int k = 0; k < 32; ++k) s += as_[r * 32 + k] * WgA[(32 + k) * 256 + c];
        asp[j] = s;
    } else if (idx < 18 * 256) {
        const int j = idx - 15 * 256, r = j >> 8, c = j & 255;
        float s = bgP[c];
        for (int k = 0; k < 32; ++k) s += ph[r * 32 + k] * WgP[k * 256 + c];
        phg[j] = 1.0f / (1.0f + expf(-s));
    }
}

// enr hidden layer: h[e,c] = gelu_exact(x[e,0:3] @ Wg1 + bg1)
__global__ void enr_h_kernel(const float* __restrict__ xe, const float* __restrict__ Wg1,
                             const float* __restrict__ bg1, float* __restrict__ h, int E_)
{
    const long long idx = (long long)blockIdx.x * blockDim.x + threadIdx.x;
    if (idx >= (long long)E_ * 256) return;
    const int e = (int)(idx >> 8), c = (int)(idx & 255);
    const float* x = xe + (size_t)e * 3;
    float s = bg1[c] + x[0] * Wg1[c] + x[1] * Wg1[256 + c] + x[2] * Wg1[512 + c];
    h[idx] = 0.5f * s * (1.0f + erff(s * 0.70710678118654752f));
}

__global__ void fill_f32(float* __restrict__ p, size_t n, float v)
{
    size_t i = (size_t)blockIdx.x * blockDim.x + threadIdx.x;
    const size_t stride = (size_t)gridDim.x * blockDim.x;
    for (; i < n; i += stride) p[i] = v;
}

__device__ __forceinline__ void atomic_max_f32(float* addr, float v)
{
    // sign-split trick; buffer initialized to -inf
    if (v >= 0.0f) atomicMax((int*)addr, __float_as_int(v));
    else           atomicMin((unsigned int*)addr, __float_as_uint(v));
}

// ---------------------------------------------------------------------------
// Edge pass 1: gather transformed feats (b128), apply gate, store message,
//              per-head logits (8-lane shuffle reduce), atomic max per (dst,h).
// gmode: 0=none(rev) 1=affects(tables) 2=pheno(table) 3=enr(buffer)
// Lane owns 8 contiguous channels; head = lane>>3 (HD=64, HEADS=4).
// ---------------------------------------------------------------------------
__global__ void __launch_bounds__(256)
edge_pass1(const int* __restrict__ ei, int E_,
           const float* __restrict__ TS, const float* __restrict__ TD,
           const float* __restrict__ attn, int gmode,
           const int* __restrict__ ea,
           const float* __restrict__ gtabA, const float* __restrict__ gtabB,
           const float* __restrict__ gbias, const float* __restrict__ gbuf,
           float* __restrict__ msg, float* __restrict__ logits,
           float* __restrict__ mmax)
{
    const int e = blockIdx.x * 8 + (threadIdx.x >> 5);
    if (e >= E_) return;
    const int lane = threadIdx.x & 31;
    const int src = ei[e];
    const int dst = ei[E_ + e];
    const int c0 = lane * 8;

    float sarr[8], darr[8], mv[8];
    {
        const float* ps = TS + (size_t)src * CCH + c0;
        const float* pd = TD + (size_t)dst * CCH + c0;
        *(float4*)(&sarr[0]) = *(const float4*)(ps);
        *(float4*)(&sarr[4]) = *(const float4*)(ps + 4);
        *(float4*)(&darr[0]) = *(const float4*)(pd);
        *(float4*)(&darr[4]) = *(const float4*)(pd + 4);
    }

    float garr[8];
#pragma unroll
    for (int i = 0; i < 8; ++i) garr[i] = 1.0f;
    if (gmode == 1) {
        const int a0 = ea[e * 2], a1 = ea[e * 2 + 1];
        float ta[8], tb[8], bb[8];
        *(float4*)(&ta[0]) = *(const float4*)(gtabA + a0 * 256 + c0);
        *(float4*)(&ta[4]) = *(const float4*)(gtabA + a0 * 256 + c0 + 4);
        *(float4*)(&tb[0]) = *(const float4*)(gtabB + a1 * 256 + c0);
        *(float4*)(&tb[4]) = *(const float4*)(gtabB + a1 * 256 + c0 + 4);
        *(float4*)(&bb[0]) = *(const float4*)(gbias + c0);
        *(float4*)(&bb[4]) = *(const float4*)(gbias + c0 + 4);
#pragma unroll
        for (int i = 0; i < 8; ++i)
            garr[i] = 1.0f / (1.0f + expf(-(ta[i] + tb[i] + bb[i])));
    } else if (gmode == 2) {
        const int a0 = ea[e];
        *(float4*)(&garr[0]) = *(const float4*)(gtabA + a0 * 256 + c0);
        *(float4*)(&garr[4]) = *(const float4*)(gtabA + a0 * 256 + c0 + 4);
    } else if (gmode == 3) {
        const float* pg = gbuf + (size_t)e * CCH + c0;
        *(float4*)(&garr[0]) = *(const float4*)(pg);
        *(float4*)(&garr[4]) = *(const float4*)(pg + 4);
    }

    float aarr[8];
    *(float4*)(&aarr[0]) = *(const float4*)(attn + c0);
    *(float4*)(&aarr[4]) = *(const float4*)(attn + c0 + 4);

    float p = 0.0f;
#pragma unroll
    for (int i = 0; i < 8; ++i) {
        float m = (sarr[i] + darr[i]) * garr[i];
        mv[i] = m;
        p += sarr[i] * darr[i] * 0.125f + m * aarr[i];  // HD^-0.5 = 1/8
    }
    {
        float* pm = msg + (size_t)e * CCH + c0;
        *(float4*)(pm)     = *(const float4*)(&mv[0]);
        *(float4*)(pm + 4) = *(const float4*)(&mv[4]);
    }

    // reduce logit over the 8 lanes of each head group
    p += __shfl_xor(p, 1, 32);
    p += __shfl_xor(p, 2, 32);
    p += __shfl_xor(p, 4, 32);
    if ((lane & 7) == 0) {
        const int h = lane >> 3;
        logits[(size_t)e * 4 + h] = p;
        atomic_max_f32(mmax + (size_t)dst * 4 + h, p);
    }
}

// Edge pass 2: ex = exp(l - max); accumulate per-(dst,head) sums
__global__ void edge_pass2(const int* __restrict__ ei, int E_,
                           const float* __restrict__ mmax, float* __restrict__ logits,
                           float* __restrict__ ssum)
{
    const int idx = blockIdx.x * blockDim.x + threadIdx.x;
    if (idx >= E_ * 4) return;
    const int e = idx >> 2, h = idx & 3;
    const int dst = ei[E_ + e];
    const float ex = expf(logits[idx] - mmax[(size_t)dst * 4 + h]);
    logits[idx] = ex;
    atomicAdd(ssum + (size_t)dst * 4 + h, ex);
}

// Edge pass 3: weighted scatter-add of messages into dst accumulators
__global__ void __launch_bounds__(256)
edge_pass3(const int* __restrict__ ei, int E_,
           const float* __restrict__ msg, const float* __restrict__ logits,
           const float* __restrict__ ssum, float* __restrict__ agg)
{
    const int e = blockIdx.x * 8 + (threadIdx.x >> 5);
    if (e >= E_) return;
    const int lane = threadIdx.x & 31;
    const int dst = ei[E_ + e];
    const int h = lane >> 3;
    const float w = logits[(size_t)e * 4 + h] / (ssum[(size_t)dst * 4 + h] + 1e-16f);
    const int c0 = lane * 8;
    float pm[8];
    {
        const float* mp = msg + (size_t)e * CCH + c0;
        *(float4*)(&pm[0]) = *(const float4*)(mp);
        *(float4*)(&pm[4]) = *(const float4*)(mp + 4);
    }
    float* pa = agg + (size_t)dst * CCH + c0;
#pragma unroll
    for (int i = 0; i < 8; ++i) atomicAdd(pa + i, pm[i] * w);
}

// ---------------------------------------------------------------------------
extern "C" void kernel_launch(void* const* d_in, const int* in_sizes, int n_in,
                              void* d_out, int out_size, void* d_ws, size_t ws_size,
                              hipStream_t stream)
{
    (void)n_in; (void)out_size; (void)ws_size;
#define FP(i) ((const float*)d_in[i])
#define IP(i) ((const int*)d_in[i])
    // setup_inputs() insertion order: x(4), ei(4), ea_int(2), ea_cont(1), params...
    const float* x_chem = FP(0);
    const float* x_gene = FP(1);
    const float* x_go   = FP(2);
    const float* x_path = FP(3);
    const int E_  = in_sizes[4] / 2;
    const int Nc  = in_sizes[0] / CCH;
    const int Ng  = in_sizes[1] / CCH;
    const int Ngo = in_sizes[2] / CCH;
    const int Np  = in_sizes[3] / CCH;

    float* ws  = (float*)d_ws;
    float* out = (float*)d_out;

    // workspace layout (floats); Nc = largest node count
    const size_t off_ts  = 0;
    const size_t off_td  = off_ts  + (size_t)Nc * CCH;
    const size_t off_agg = off_td  + (size_t)Nc * CCH;
    const size_t off_msg = off_agg + (size_t)Nc * CCH;     // also enr hidden buffer
    const size_t off_gat = off_msg + (size_t)E_ * CCH;     // enr gate
    const size_t off_log = off_gat + (size_t)E_ * CCH;
    const size_t off_m   = off_log + (size_t)E_ * 4;
    const size_t off_s   = off_m   + (size_t)Nc * 4;
    const size_t off_tab = off_s   + (size_t)Nc * 4;
    float* TS   = ws + off_ts;
    float* TD   = ws + off_td;
    float* AGG  = ws + off_agg;
    float* MSG  = ws + off_msg;
    float* GATE = ws + off_gat;
    float* LOG  = ws + off_log;
    float* MM   = ws + off_m;
    float* SS   = ws + off_s;
    float* ATP  = ws + off_tab;            // [10,256]
    float* ASP  = ATP + 10 * 256;          // [5,256]
    float* PHG  = ASP + 5 * 256;           // [3,256] (pre-sigmoided)

    // categorical gate tables: emb.at=39, emb.as=40, emb.ph=41;
    // affects.Wg=16 ; pheno.Wg=23, pheno.bg=24
    gate_tables<<<18, 256, 0, stream>>>(FP(39), FP(40), FP(41), FP(16), FP(23), FP(24),
                                        ATP, ASP, PHG);

    struct TypeDesc {
        const float* xs; int ns; const float* xd; int nd;
        const int* ei;
        const float *Ws, *bs, *Wd, *bd, *attn;
        int gmode; const int* ea;
        const float *Wout, *bout; size_t out_off;
    };
    const TypeDesc T[4] = {
        // affects: chem -> gene
        { x_chem, Nc, x_gene, Ng, IP(4), FP(11), FP(12), FP(13), FP(14), FP(15),
          1, IP(8), FP(44), FP(45), (size_t)Nc * CCH },
        // pheno: chem -> go_term
        { x_chem, Nc, x_go, Ngo, IP(5), FP(18), FP(19), FP(20), FP(21), FP(22),
          2, IP(9), FP(46), FP(47), (size_t)(Nc + Ng) * CCH },
        // enr: chem -> pathway
        { x_chem, Nc, x_path, Np, IP(6), FP(25), FP(26), FP(27), FP(28), FP(29),
          3, nullptr, FP(48), FP(49), (size_t)(Nc + Ng + Ngo) * CCH },
        // rev: gene -> chem
        { x_gene, Ng, x_chem, Nc, IP(7), FP(34), FP(35), FP(36), FP(37), FP(38),
          0, nullptr, FP(42), FP(43), 0 },
    };

    for (int t = 0; t < 4; ++t) {
        const TypeDesc& td = T[t];
        // node transforms (WMMA GEMMs, 32x64 per wave, B panel in LDS)
        dim3 gs((td.ns + 127) / 128, 4);
        wmma_gemm_bias<<<gs, 128, 0, stream>>>(td.xs, td.Ws, td.bs, TS, td.ns, 0);
        dim3 gd((td.nd + 127) / 128, 4);
        wmma_gemm_bias<<<gd, 128, 0, stream>>>(td.xd, td.Wd, td.bd, TD, td.nd, 0);

        if (td.gmode == 3) {
            // enr gate MLP: h = gelu(x@Wg1+bg1) (VALU), gate = sigmoid(h@Wg2+bg2) (WMMA)
            const long long tot = (long long)E_ * 256;
            enr_h_kernel<<<(int)((tot + 255) / 256), 256, 0, stream>>>(
                FP(10), FP(30), FP(31), MSG, E_);
            dim3 gg((E_ + 127) / 128, 4);
            wmma_gemm_bias<<<gg, 128, 0, stream>>>(MSG, FP(32), FP(33), GATE, E_, 1);
        }

        fill_f32<<<64, 256, 0, stream>>>(MM, (size_t)td.nd * 4, -INFINITY);
        fill_f32<<<64, 256, 0, stream>>>(SS, (size_t)td.nd * 4, 0.0f);
        fill_f32<<<1024, 256, 0, stream>>>(AGG, (size_t)td.nd * CCH, 0.0f);

        edge_pass1<<<(E_ + 7) / 8, 256, 0, stream>>>(
            td.ei, E_, TS, TD, td.attn, td.gmode, td.ea,
            td.gmode == 1 ? ATP : (td.gmode == 2 ? PHG : nullptr),
            td.gmode == 1 ? ASP : nullptr,
            td.gmode == 1 ? FP(17) : nullptr,
            td.gmode == 3 ? GATE : nullptr,
            MSG, LOG, MM);
        edge_pass2<<<(E_ * 4 + 255) / 256, 256, 0, stream>>>(td.ei, E_, MM, LOG, SS);
        edge_pass3<<<(E_ + 7) / 8, 256, 0, stream>>>(td.ei, E_, MSG, LOG, SS, AGG);

        // output projection straight into d_out slice (WMMA GEMM)
        dim3 go((td.nd + 127) / 128, 4);
        wmma_gemm_bias<<<go, 128, 0, stream>>>(AGG, td.Wout, td.bout,
                                               out + td.out_off, td.nd, 0);
    }
#undef FP
#undef IP
}